// FanProj2D_29850022707685
// MI455X (gfx1250) — compile-verified
//
#include <hip/hip_runtime.h>
#include <hip/hip_bf16.h>
#include <math.h>

// ---------------- geometry constants (match reference) ----------------
#define VOL    256
#define NPROJ  256
#define NDET   384
#define NS     256

#define BLOCK_THREADS 512
#define RAYS_TOTAL (2 * NPROJ * NDET)           // 196608
#define NBLOCKS    (RAYS_TOTAL / BLOCK_THREADS) // 384

// LDS layout: [0,260) zero pad | [260, 260+65536) image | 260-float zero pad
#define LDS_BIAS   260                               // 1040 B, multiple of 16
#define LDS_FLOATS (VOL * VOL + 2 * LDS_BIAS)        // 66056
#define LDS_BYTES  (LDS_FLOATS * sizeof(float))      // 264224 B < 320 KB

// derived scalars:
// diag = 256*sqrt(2); dt = diag/256 = sqrt(2); t0 = SID - diag/2 + 0.5*dt
#define DT_F   1.4142135623730951f
#define T0_F   219.68777079742835f
#define TWOPI_OVER_P 0.024543692606170259f   // 2*pi/256
#define SDD_F  800.0f
#define SDD2_F 640000.0f
#define DET_SPACING_F 1.2f
#define DET_CENTER_F  191.5f                  // (NDET-1)/2
#define HALF_VOL_F    127.5f                  // (VOL-1)/2

// async-to-LDS builtin pointer types (int4* in AS1/AS3 per hipcc signature)
typedef int v4i __attribute__((vector_size(4 * sizeof(int))));
typedef __attribute__((address_space(1))) v4i gv4i_t;
typedef __attribute__((address_space(3))) v4i lv4i_t;

__global__ __launch_bounds__(BLOCK_THREADS)
void fanproj2d_kernel(const float* __restrict__ img, float* __restrict__ out)
{
    extern __shared__ float lds[];

    const int tid  = threadIdx.x;
    const int ray0 = blockIdx.x * BLOCK_THREADS;
    const int b    = ray0 / (NPROJ * NDET);    // 0 or 1; each block owns one batch image
    const float* __restrict__ ibase = img + (size_t)b * VOL * VOL;

    // ---- zero guard pads (disjoint from async-staged region) ----
    if (tid < LDS_BIAS) {
        lds[tid] = 0.0f;                                  // front pad
        lds[LDS_BIAS + VOL * VOL + tid] = 0.0f;           // back pad
    }

    // ---- stage whole image into LDS: 65536 floats = 16384 x b128, 32 per thread ----
    {
        const char* g = (const char*)ibase;
        char*       l = (char*)lds + LDS_BIAS * sizeof(float);  // 1040 B, 16B aligned
#if __has_builtin(__builtin_amdgcn_global_load_async_to_lds_b128)
        #pragma unroll
        for (int i = 0; i < 32; ++i) {
            const int off = (tid + i * BLOCK_THREADS) * 16;     // coalesced 16B chunks
            __builtin_amdgcn_global_load_async_to_lds_b128(
                (gv4i_t*)(g + off), (lv4i_t*)(l + off), 0, 0);
        }
#if __has_builtin(__builtin_amdgcn_s_wait_asynccnt)
        __builtin_amdgcn_s_wait_asynccnt(0);
#else
        asm volatile("s_wait_asynccnt 0" ::: "memory");
#endif
#else
        // fallback: vector copy through VGPRs
        #pragma unroll
        for (int i = 0; i < 32; ++i) {
            const int e4 = tid + i * BLOCK_THREADS;             // float4 index
            ((float4*)l)[e4] = ((const float4*)g)[e4];
        }
#endif
    }
    __syncthreads();   // pads stored + all 16 waves' staged data visible

    // ---- per-thread ray geometry (registers only) ----
    const int ray = ray0 + tid;
    const int rem = ray - b * (NPROJ * NDET);
    const int p   = rem / NDET;
    const int d   = rem - p * NDET;

    float sn, cs;
    __sincosf((float)p * TWOPI_OVER_P, &sn, &cs);

    const float srcx = -400.0f * cs;
    const float srcy = -400.0f * sn;
    const float doff = ((float)d - DET_CENTER_F) * DET_SPACING_F;
    float rdx = SDD_F * cs - doff * sn;        // SDD*cen + doff*u, u = (-sin, cos)
    float rdy = SDD_F * sn + doff * cs;
    const float rn = rsqrtf(SDD2_F + doff * doff);  // |det-src| = sqrt(SDD^2 + doff^2)
    rdx *= rn;
    rdy *= rn;

    // marching point in fractional pixel coords, advanced incrementally
    float px = srcx + T0_F * rdx + HALF_VOL_F;
    float py = srcy + T0_F * rdy + HALF_VOL_F;
    const float sx = DT_F * rdx;
    const float sy = DT_F * rdy;

    float acc = 0.0f;
    #pragma unroll 4
    for (int s = 0; s < NS; ++s) {
        const float xf = floorf(px);
        const float yf = floorf(py);
        const float wx = px - xf;
        const float wy = py - yf;
        const float ox = 1.0f - wx;
        const float oy = 1.0f - wy;
        const int x0 = (int)xf;
        const int y0 = (int)yf;
        const int x1 = x0 + 1;
        const int y1 = y0 + 1;

        // bounds folded into weights: zero weight annihilates any finite tap value
        const float oxe = ((unsigned)x0 < (unsigned)VOL) ? ox : 0.0f;
        const float wxe = ((unsigned)x1 < (unsigned)VOL) ? wx : 0.0f;
        const float oye = ((unsigned)y0 < (unsigned)VOL) ? oy : 0.0f;
        const float wye = ((unsigned)y1 < (unsigned)VOL) ? wy : 0.0f;

        // single clamped base; 4 taps at immediate offsets 0, +1, +256, +257.
        // clamp to [-1,255] keeps neighbor taps address-aligned with their weights;
        // guard rows land in the zeroed pads. idx range: [3, 66052] -- always in LDS.
        const int xc = min(max(x0, -1), VOL - 1);
        const int yc = min(max(y0, -1), VOL - 1);
        const int idx = LDS_BIAS + (yc << 8) + xc;

        const float v00 = lds[idx];
        const float v01 = lds[idx + 1];
        const float v10 = lds[idx + VOL];
        const float v11 = lds[idx + VOL + 1];

        acc += oye * (oxe * v00 + wxe * v01) + wye * (oxe * v10 + wxe * v11);

        px += sx;
        py += sy;
    }

    out[ray] = acc * DT_F;
}

extern "C" void kernel_launch(void* const* d_in, const int* in_sizes, int n_in,
                              void* d_out, int out_size, void* d_ws, size_t ws_size,
                              hipStream_t stream)
{
    (void)in_sizes; (void)n_in; (void)out_size; (void)d_ws; (void)ws_size;
    const float* x = (const float*)d_in[0];   // [2,256,256,1] fp32
    float* out     = (float*)d_out;           // [2,256,384,1] fp32

    hipLaunchKernelGGL(fanproj2d_kernel,
                       dim3(NBLOCKS), dim3(BLOCK_THREADS),
                       LDS_BYTES, stream,
                       x, out);
}